// DeformConv3d_60919816126926
// MI455X (gfx1250) — compile-verified
//
#include <hip/hip_runtime.h>

// MI455X / gfx1250, wave32. All matrix math via V_WMMA_F32_16X16X4_F32 (full fp32,
// since the problem is HBM-bound: ~60MB moved vs ~13 GFLOP).

typedef __attribute__((ext_vector_type(2))) float v2f;
typedef __attribute__((ext_vector_type(4))) float v4f;
typedef __attribute__((ext_vector_type(8))) float v8f;

#define HH 128
#define WW 128
#define CC 64
#define HP 130   // padded (PAD=1)
#define WP 130
#define NCH 99   // 18*5 + 9 conv output channels, padded to 128
#define CSTR 584 // central row stride in dwords (576 padded, keeps 8B align, spreads banks)

// ---------------------------------------------------------------------------
// Kernel 1: pack weights/biases into WMMA-fragment-friendly layouts.
//  Bp2  : [nt=8][kp=800][n=16][2]  k = tap*64 + ic, tap = ky*5+kx   (204800 f)
//  bias : [128]                                                     (128 f)
//  Wp2  : [nt=4][kp=288][n=16][2]  k = tap*64 + c,  tap = 3x3 idx   (36864 f)
// ---------------------------------------------------------------------------
__global__ __launch_bounds__(256) void pack_weights(
    const float* __restrict__ wp1, const float* __restrict__ bp1,
    const float* __restrict__ wp2, const float* __restrict__ bp2,
    const float* __restrict__ wp4, const float* __restrict__ bp4,
    const float* __restrict__ wp5, const float* __restrict__ bp5,
    const float* __restrict__ wxy, const float* __restrict__ bxy,
    const float* __restrict__ wh,  const float* __restrict__ bh,
    const float* __restrict__ wconv,
    float* __restrict__ Bp2, float* __restrict__ biasAll, float* __restrict__ Wp2)
{
    int idx = blockIdx.x * 256 + threadIdx.x;
    if (idx < 204800) {
        int h  = idx & 1;
        int n  = (idx >> 1) & 15;
        int t2 = idx >> 5;
        int kp = t2 % 800;
        int nt = t2 / 800;
        int k  = kp * 2 + h;
        int tap = k >> 6, ic = k & 63;
        int rr = tap / 5, cc = tap % 5;
        int oc = nt * 16 + n;
        float v = 0.f;
        const float* ws = nullptr; int o = 0;
        if      (oc < 18) { ws = wp1; o = oc; }
        else if (oc < 36) { ws = wp2; o = oc - 18; }
        else if (oc < 54) { ws = wp4; o = oc - 36; }
        else if (oc < 72) { ws = wp5; o = oc - 54; }
        else if (oc < 90) { ws = wxy; o = oc - 72; }
        else if (oc < 99) { ws = wh;  o = oc - 90; }
        if (ws) v = ws[((o * 64 + ic) * 5 + rr) * 5 + cc];
        Bp2[idx] = v;
    } else if (idx < 204800 + 128) {
        int oc = idx - 204800;
        float v = 0.f;
        if      (oc < 18) v = bp1[oc];
        else if (oc < 36) v = bp2[oc - 18];
        else if (oc < 54) v = bp4[oc - 36];
        else if (oc < 72) v = bp5[oc - 54];
        else if (oc < 90) v = bxy[oc - 72];
        else if (oc < 99) v = bh[oc - 90];
        biasAll[oc] = v;
    } else if (idx < 204800 + 128 + 36864) {
        int j  = idx - 204928;
        int h  = j & 1;
        int n  = (j >> 1) & 15;
        int t2 = j >> 5;
        int kp = t2 % 288;
        int nt = t2 / 288;
        int k  = kp * 2 + h;
        int tap = k >> 6, c = k & 63;
        int oc = nt * 16 + n;
        Wp2[j] = wconv[(oc * 64 + c) * 9 + tap];
    }
}

// ---------------------------------------------------------------------------
// Kernel 2: pad (+1 each side, zeros) and transpose images to NHWC so each
// bilinear tap gathers 64 contiguous floats (256B). IMG[t][b][X][Y][c].
// ---------------------------------------------------------------------------
__global__ __launch_bounds__(256) void pack_images(
    const float* __restrict__ i1, const float* __restrict__ i2,
    const float* __restrict__ i4, const float* __restrict__ i5,
    float* __restrict__ IMG)
{
    long long idx = (long long)blockIdx.x * 256 + threadIdx.x;
    const long long total = 4LL * 2 * HP * WP * CC;
    if (idx >= total) return;
    int Y = (int)(idx % WP); long long t2 = idx / WP;
    int X = (int)(t2 % HP); t2 /= HP;
    int c = (int)(t2 & 63); t2 >>= 6;
    int b = (int)(t2 & 1);
    int t = (int)(t2 >> 1);
    const float* src = (t == 0) ? i1 : (t == 1) ? i2 : (t == 2) ? i4 : i5;
    float v = 0.f;
    if (X >= 1 && X <= HH && Y >= 1 && Y <= WW)
        v = src[((b * CC + c) * HH + (X - 1)) * WW + (Y - 1)];
    IMG[((((long long)t * 2 + b) * HP + X) * WP + Y) * CC + c] = v;
}

// ---------------------------------------------------------------------------
// Kernel 3: fused 5x5 convs as implicit GEMM with V_WMMA_F32_16X16X4_F32.
// M-tile = 16 pixels of one row; 8 waves = 8 N-tiles (128 chans, 99 real).
// K = 1600 (tap-major, ic chunks of 4). Output channel-last: OFF[pix][128].
// ---------------------------------------------------------------------------
__global__ __launch_bounds__(256) void conv5x5_wmma(
    const float* __restrict__ src, const float* __restrict__ Bp2,
    const float* __restrict__ biasAll, float* __restrict__ OFF)
{
    __shared__ float patch[5 * 20 * CC];   // [row][col][ic], 25.6 KB
    int tile = blockIdx.x;
    int b  = tile >> 10;
    int y  = (tile & 1023) >> 3;
    int x0 = (tile & 7) << 4;
    int tid = threadIdx.x;

    for (int e = tid; e < 5 * 20 * CC; e += 256) {
        int ic = e & 63;
        int cc = (e >> 6) % 20;
        int rr = (e >> 6) / 20;
        int gy = y + rr - 2;
        int gx = x0 + cc - 2;
        float v = 0.f;
        if (gy >= 0 && gy < HH && gx >= 0 && gx < WW)
            v = src[((b * CC + ic) * HH + gy) * WW + gx];
        patch[e] = v;
    }
    __syncthreads();

    int wv = tid >> 5;
    int L  = tid & 31;
    int mn = L & 15;               // M for A / N for B (same lane slot)
    int koff = (L >> 4) << 1;      // lanes 0-15: K 0,1 ; lanes 16-31: K 2,3

    v8f acc = {};
    const float* bbase = Bp2 + ((size_t)wv * 800 + (koff >> 1)) * 32 + mn * 2;
    for (int tap = 0; tap < 25; ++tap) {
        int rr = tap / 5, kx = tap % 5;
        const float* pa = patch + ((rr * 20) + kx + mn) * CC + koff;
        const float* pb = bbase + tap * 32 * 32;
#pragma unroll
        for (int icc = 0; icc < 16; ++icc) {
            v2f a  = *(const v2f*)(pa + icc * 4);     // A[m][k..k+1] from LDS
            v2f bf = *(const v2f*)(pb + icc * 64);    // B[k..k+1][n] packed pairs
            acc = __builtin_amdgcn_wmma_f32_16x16x4_f32(
                false, a, false, bf, (short)0, acc, false, false);
        }
    }

    float bs = biasAll[wv * 16 + mn];
    int pbase = tile * 16;
    int mh = (L >> 4) * 8;         // C/D: VGPR r -> M = r (+8 for hi lanes)
#pragma unroll
    for (int r = 0; r < 8; ++r)
        OFF[(size_t)(pbase + r + mh) * 128 + wv * 16 + mn] = acc[r] + bs;
}

// ---------------------------------------------------------------------------
// Kernel 4: fused get_p + bilinear sample (4 imgs) + IDW blend + final 64x576
// contraction (WMMA). central[16][576] lives only in LDS.
// ---------------------------------------------------------------------------
__global__ __launch_bounds__(256) void fuse_sample_wmma(
    const float* __restrict__ OFF, const float* __restrict__ IMG,
    const float* __restrict__ Wp2, float* __restrict__ out)
{
    __shared__ float smem[1600 + 16 * CSTR];   // 43.8 KB
    float* off_lds = smem;                     // [16][100]
    float* cen_lds = smem + 1600;              // [16][CSTR]
    float* out_lds = smem;                     // alias (used after off is dead)

    int tile = blockIdx.x;
    int b  = tile >> 10;
    int y  = (tile & 1023) >> 3;
    int x0 = (tile & 7) << 4;
    int tid = threadIdx.x;

    for (int e = tid; e < 16 * 100; e += 256) {
        int m = e / 100, ch = e % 100;
        off_lds[e] = (ch < NCH) ? OFF[(size_t)(tile * 16 + m) * 128 + ch] : 0.f;
    }
    __syncthreads();

    int wv = tid >> 5, L = tid & 31;
    int c0 = L << 1;                           // this lane owns channels c0,c0+1
    const float zs[4] = {0.f, 1.f / 3.f, 2.f / 3.f, 1.f};

    for (int pi = 0; pi < 18; ++pi) {          // 8 waves x 18 = 144 (m,n) pairs
        int pr = wv * 18 + pi;
        int m = pr / 9, n = pr % 9;
        const float* om = off_lds + m * 100;
        float pnx = (float)(n / 3 - 1), pny = (float)(n % 3 - 1);
        float bx = pnx + (float)(y + 1);
        float by = pny + (float)(x0 + m + 1);
        // IDW reference point (h_xy, h_z)
        float hx = fminf(fmaxf(om[72 + n] + bx, 0.f), 129.f);
        float hy = fminf(fmaxf(om[81 + n] + by, 0.f), 129.f);
        float hz = fminf(fmaxf(om[90 + n] + 1.5f, 0.f), 3.f) * (1.f / 3.f);

        float cen0 = 0.f, cen1 = 0.f, den = 0.f;
#pragma unroll
        for (int t = 0; t < 4; ++t) {
            float px = om[t * 18 + n] + bx;
            float py = om[t * 18 + 9 + n] + by;
            float fx = floorf(px), fy = floorf(py);
            float ltx = fminf(fmaxf(fx, 0.f), 129.f);
            float lty = fminf(fmaxf(fy, 0.f), 129.f);
            float rbx = fminf(fmaxf(fx + 1.f, 0.f), 129.f);
            float rby = fminf(fmaxf(fy + 1.f, 0.f), 129.f);
            float pcx = fminf(fmaxf(px, 0.f), 129.f);
            float pcy = fminf(fmaxf(py, 0.f), 129.f);
            float glt = (1.f + (ltx - pcx)) * (1.f + (lty - pcy));
            float grb = (1.f - (rbx - pcx)) * (1.f - (rby - pcy));
            float glb = (1.f - (rbx - pcx)) * (1.f + (lty - pcy));
            float grt = (1.f + (ltx - pcx)) * (1.f - (rby - pcy));
            int ilx = (int)ltx, ily = (int)lty, irx = (int)rbx, iry = (int)rby;
            const float* base = IMG + (size_t)(t * 2 + b) * HP * WP * CC + c0;
            v2f vlt = *(const v2f*)(base + (size_t)(ilx * WP + ily) * CC);
            v2f vrb = *(const v2f*)(base + (size_t)(irx * WP + iry) * CC);
            v2f vlb = *(const v2f*)(base + (size_t)(irx * WP + ily) * CC);
            v2f vrt = *(const v2f*)(base + (size_t)(ilx * WP + iry) * CC);
            float f0 = glt * vlt.x + grb * vrb.x + glb * vlb.x + grt * vrt.x;
            float f1 = glt * vlt.y + grb * vrb.y + glb * vlb.y + grt * vrt.y;
            float ddx = (hx - pcx) * (1.f / 129.f);
            float ddy = (hy - pcy) * (1.f / 129.f);
            float ddz = hz - zs[t];
            float wt = 1.f / (ddx * ddx + ddy * ddy + ddz * ddz + 1e-15f);
            cen0 += wt * f0; cen1 += wt * f1; den += wt;
        }
        float inv = 1.f / den;
        v2f cv; cv.x = cen0 * inv; cv.y = cen1 * inv;
        *(v2f*)(cen_lds + m * CSTR + n * 64 + c0) = cv;   // K = n*64 + c
    }
    __syncthreads();

    if (wv < 4) {                  // contraction: out[m][oc] = W[oc][k] central[m][k]
        int mn = L & 15;
        int koff = (L >> 4) << 1;
        v8f acc = {};
        const float* wbase = Wp2 + ((size_t)wv * 288 + (koff >> 1)) * 32 + mn * 2;
        for (int ntap = 0; ntap < 9; ++ntap) {
            const float* pa = cen_lds + mn * CSTR + ntap * 64 + koff;
            const float* pb = wbase + ntap * 32 * 32;
#pragma unroll
            for (int icc = 0; icc < 16; ++icc) {
                v2f a  = *(const v2f*)(pa + icc * 4);
                v2f bf = *(const v2f*)(pb + icc * 64);
                acc = __builtin_amdgcn_wmma_f32_16x16x4_f32(
                    false, a, false, bf, (short)0, acc, false, false);
            }
        }
        int mh = (L >> 4) * 8;
#pragma unroll
        for (int r = 0; r < 8; ++r)
            out_lds[(wv * 16 + mn) * 17 + (r + mh)] = acc[r]; // [oc][m], pad 17
    }
    __syncthreads();

    {   // coalesced store: 4 threads emit one 64B row per output channel
        int oc = tid >> 2, q = tid & 3;
        const float* po = out_lds + oc * 17 + q * 4;
        v4f vv; vv.x = po[0]; vv.y = po[1]; vv.z = po[2]; vv.w = po[3];
        size_t oidx = (((size_t)b * 64 + oc) * HH + y) * WW + x0 + q * 4;
        *(v4f*)(out + oidx) = vv;
    }
}

// ---------------------------------------------------------------------------
extern "C" void kernel_launch(void* const* d_in, const int* in_sizes, int n_in,
                              void* d_out, int out_size, void* d_ws, size_t ws_size,
                              hipStream_t stream)
{
    (void)in_sizes; (void)n_in; (void)out_size; (void)ws_size;
    const float* img1 = (const float*)d_in[0];
    const float* img2 = (const float*)d_in[1];
    const float* img4 = (const float*)d_in[2];
    const float* img5 = (const float*)d_in[3];
    const float* osrc = (const float*)d_in[4];
    const float* wp1 = (const float*)d_in[5];  const float* bp1 = (const float*)d_in[6];
    const float* wp2 = (const float*)d_in[7];  const float* bp2 = (const float*)d_in[8];
    const float* wp4 = (const float*)d_in[9];  const float* bp4 = (const float*)d_in[10];
    const float* wp5 = (const float*)d_in[11]; const float* bp5 = (const float*)d_in[12];
    const float* wxy = (const float*)d_in[13]; const float* bxy = (const float*)d_in[14];
    const float* wh  = (const float*)d_in[15]; const float* bh  = (const float*)d_in[16];
    const float* wconv = (const float*)d_in[17];

    float* ws   = (float*)d_ws;                // ~52.4 MB used
    float* OFF  = ws;                          // 4,194,304 f : (B*H*W, 128) ch-last
    float* IMG  = OFF + 4194304;               // 8,652,800 f : NHWC padded x4 imgs
    float* Bp2  = IMG + 8652800;               //   204,800 f
    float* biasAll = Bp2 + 204800;             //       128 f
    float* Wp2  = biasAll + 128;               //    36,864 f

    pack_weights<<<(204800 + 128 + 36864 + 255) / 256, 256, 0, stream>>>(
        wp1, bp1, wp2, bp2, wp4, bp4, wp5, bp5, wxy, bxy, wh, bh, wconv,
        Bp2, biasAll, Wp2);
    pack_images<<<(int)((4LL * 2 * HP * WP * CC + 255) / 256), 256, 0, stream>>>(
        img1, img2, img4, img5, IMG);
    conv5x5_wmma<<<2048, 256, 0, stream>>>(osrc, Bp2, biasAll, OFF);
    fuse_sample_wmma<<<2048, 256, 0, stream>>>(OFF, IMG, Wp2, (float*)d_out);
}